// GCNBlock_39565238731081
// MI455X (gfx1250) — compile-verified
//
#include <hip/hip_runtime.h>
#include <hip/hip_bf16.h>

// Problem constants (match reference)
#define NNODES 40000
#define DFEAT  128
#define NEDGES 640000
#define GALPHA 0.1f
#define BN_EPS 1e-5f

typedef __attribute__((ext_vector_type(2))) float v2f;
typedef __attribute__((ext_vector_type(8))) float v8f;

// Workspace layout (in floats):
//   [0, NP)                 : deg, overwritten in place with dinv
//   [NP, NP + N*D)          : agg
//   [STAT_OFF + 0*D .. 1*D) : colsum
//   [STAT_OFF + 1*D .. 2*D) : colsumsq
//   [STAT_OFF + 2*D .. 3*D) : scale
//   [STAT_OFF + 3*D .. 4*D) : shift
static constexpr long long NP       = 40064;                       // padded node count
static constexpr long long AGG_OFF  = NP;
static constexpr long long STAT_OFF = NP + (long long)NNODES * DFEAT;

// LDS row strides (floats). Both even (keeps b64 loads 8B-aligned); chosen so
// fragment reads and staging writes are conflict-free on 64 banks.
#define WT_STRIDE 134   // transposed W: Wt[n][k]
#define AL_STRIDE 132   // blended A tile: Al[m][k]

// ---------------------------------------------------------------- zero scratch
__global__ void zero_kernel(float* __restrict__ p, long long n) {
    long long i = (long long)blockIdx.x * blockDim.x + threadIdx.x;
    if (i < n) p[i] = 0.0f;
}

// ---------------------------------------------------------------- degrees
// deg[col[e]] += w[e] for real edges, deg[i] += 1 for self loops
__global__ void deg_kernel(const long long* __restrict__ ei,
                           const float* __restrict__ ew,
                           float* __restrict__ deg) {
    long long i = (long long)blockIdx.x * blockDim.x + threadIdx.x;
    if (i < NEDGES) {
        long long col = ei[NEDGES + i];
        atomicAdd(&deg[col], ew[i]);
    } else if (i < NEDGES + NNODES) {
        atomicAdd(&deg[i - NEDGES], 1.0f);
    }
}

// ---------------------------------------------------------------- dinv in place
__global__ void dinv_kernel(float* __restrict__ deg) {
    long long i = (long long)blockIdx.x * blockDim.x + threadIdx.x;
    if (i < NNODES) {
        float d = deg[i];
        deg[i] = (d > 0.0f) ? rsqrtf(d) : 0.0f;
    }
}

// ---------------------------------------------------------------- aggregation
// One wave32 per (edge or self-loop); 32 lanes x 4 iters cover D=128.
// Edge scalars are lane-uniform -> scalar loads; feature traffic is L2-resident
// (x and agg are each 20.5 MB, both fit in the 192 MB L2).
__global__ __launch_bounds__(256) void agg_kernel(const float* __restrict__ x,
                                                  const long long* __restrict__ ei,
                                                  const float* __restrict__ ew,
                                                  const float* __restrict__ dinv,
                                                  float* __restrict__ agg) {
    long long wid = (long long)blockIdx.x * (blockDim.x >> 5) + (threadIdx.x >> 5);
    int lane = threadIdx.x & 31;
    if (wid >= NEDGES + NNODES) return;

    long long row, col;
    float w;
    if (wid < NEDGES) {
        row = ei[wid];
        col = ei[NEDGES + wid];
        w   = ew[wid];
    } else {
        row = col = wid - NEDGES;
        w = 1.0f;
    }
    float norm = dinv[row] * w * dinv[col];
    const float* xr = x + row * DFEAT;
    float* ac = agg + col * DFEAT;
#pragma unroll
    for (int t = 0; t < 4; ++t) {
        int d = lane + 32 * t;
        atomicAdd(&ac[d], norm * xr[d]);
    }
}

// ---------------------------------------------------------------- GEMM + ReLU + BN stats
// Block = one 16-row M tile, 8 waves = 8 N tiles of 16. fp32 WMMA 16x16x4.
// W is staged TRANSPOSED in LDS so each B fragment is a K-consecutive pair
// -> single ds_load_b64 (mergeable to 2addr_b64), matching the A path.
__global__ __launch_bounds__(256) void gemm_bn_kernel(const float* __restrict__ xorig,
                                                      const float* __restrict__ Wm,
                                                      const float* __restrict__ agg,
                                                      float* __restrict__ h,
                                                      float* __restrict__ colsum,
                                                      float* __restrict__ colsumsq) {
    __shared__ float Wt[DFEAT * WT_STRIDE];   // W^T : Wt[n*WT_STRIDE + k] (68.6 KB)
    __shared__ float Al[16 * AL_STRIDE];      // blended 16x128 A tile (8.4 KB)

    const int tid = threadIdx.x;
    const int m0  = blockIdx.x * 16;

    // Stage W transposed (coalesced global read; conflict-free LDS write)
    for (int i = tid; i < DFEAT * DFEAT; i += 256) {
        int k = i >> 7, n = i & 127;          // W row-major: W[k][n]
        Wt[n * WT_STRIDE + k] = Wm[i];
    }
    // Stage blended A tile: (1-a)*agg + a*x_orig
    for (int i = tid; i < 16 * DFEAT; i += 256) {
        int r = i >> 7, c = i & 127;
        long long g = (long long)(m0 + r) * DFEAT + c;
        Al[r * AL_STRIDE + c] = (1.0f - GALPHA) * agg[g] + GALPHA * xorig[g];
    }
    __syncthreads();

    const int wave = tid >> 5;
    const int lane = tid & 31;
    const int lm   = lane & 15;   // M (for A) / N (for B) position
    const int kh   = lane >> 4;   // K half select
    const int n0   = wave * 16;

    const float* arow = &Al[lm * AL_STRIDE];            // A row for this lane
    const float* brow = &Wt[(n0 + lm) * WT_STRIDE];     // W^T row (= W column)

    v8f acc = {};
#pragma unroll
    for (int k = 0; k < DFEAT; k += 4) {
        const int kk = k + 2 * kh;            // even -> 8B-aligned b64 loads
        // A 16x4: lanes 0-15 -> K = k,k+1 ; lanes 16-31 -> K = k+2,k+3 (M = lm)
        v2f a = *reinterpret_cast<const v2f*>(&arow[kk]);
        // B 4x16: same K split, N = n0+lm striped across lanes
        v2f b = *reinterpret_cast<const v2f*>(&brow[kk]);
        acc = __builtin_amdgcn_wmma_f32_16x16x4_f32(false, a, false, b,
                                                    (short)0, acc, false, false);
    }

    // Epilogue: ReLU, store h, accumulate per-column sum / sumsq for BatchNorm.
    // D layout: VGPR r holds M = r (lanes 0-15) / M = r+8 (lanes 16-31), N = n0+lm.
    float s = 0.0f, sq = 0.0f;
    const int n = n0 + lm;
#pragma unroll
    for (int r = 0; r < 8; ++r) {
        float v = acc[r];
        v = v > 0.0f ? v : 0.0f;
        int m = m0 + r + 8 * kh;
        h[(long long)m * DFEAT + n] = v;
        s  += v;
        sq += v * v;
    }
    // Halves of the wave cover the same column set -> combine, then 16 atomics.
    s  += __shfl_down(s, 16, 32);
    sq += __shfl_down(sq, 16, 32);
    if (kh == 0) {
        atomicAdd(&colsum[n], s);
        atomicAdd(&colsumsq[n], sq);
    }
}

// ---------------------------------------------------------------- BN finalize
__global__ void bn_finalize_kernel(const float* __restrict__ colsum,
                                   const float* __restrict__ colsumsq,
                                   const float* __restrict__ gamma,
                                   const float* __restrict__ beta,
                                   float* __restrict__ scale,
                                   float* __restrict__ shift) {
    int j = threadIdx.x;
    if (j < DFEAT) {
        float inv_n = 1.0f / (float)NNODES;
        float mean  = colsum[j] * inv_n;
        float var   = colsumsq[j] * inv_n - mean * mean;   // biased variance
        float sc    = gamma[j] * rsqrtf(var + BN_EPS);
        scale[j] = sc;
        shift[j] = beta[j] - mean * sc;
    }
}

// ---------------------------------------------------------------- BN apply (float4)
__global__ __launch_bounds__(256) void bn_apply_kernel(float* __restrict__ h,
                                                       const float* __restrict__ scale,
                                                       const float* __restrict__ shift) {
    long long i = (long long)blockIdx.x * blockDim.x + threadIdx.x;  // per float4
    const long long total = (long long)NNODES * DFEAT / 4;
    if (i >= total) return;
    int c = (int)((i * 4) & (DFEAT - 1));
    float4 v = reinterpret_cast<float4*>(h)[i];
    v.x = v.x * scale[c + 0] + shift[c + 0];
    v.y = v.y * scale[c + 1] + shift[c + 1];
    v.z = v.z * scale[c + 2] + shift[c + 2];
    v.w = v.w * scale[c + 3] + shift[c + 3];
    reinterpret_cast<float4*>(h)[i] = v;
}

// ---------------------------------------------------------------- launcher
extern "C" void kernel_launch(void* const* d_in, const int* in_sizes, int n_in,
                              void* d_out, int out_size, void* d_ws, size_t ws_size,
                              hipStream_t stream) {
    const float*     x     = (const float*)d_in[0];
    const float*     xorig = (const float*)d_in[1];
    const long long* ei    = (const long long*)d_in[2];   // (2,E) int64
    const float*     ew    = (const float*)d_in[3];
    const float*     Wm    = (const float*)d_in[4];
    const float*     gamma = (const float*)d_in[5];
    const float*     beta  = (const float*)d_in[6];
    float* out = (float*)d_out;

    float* wsf      = (float*)d_ws;
    float* deg      = wsf;                 // becomes dinv in place
    float* agg      = wsf + AGG_OFF;
    float* colsum   = wsf + STAT_OFF;
    float* colsumsq = colsum + DFEAT;
    float* scale    = colsumsq + DFEAT;
    float* shift    = scale + DFEAT;

    // 1. zero deg + agg + stats accumulators (fresh every launch)
    long long zn = STAT_OFF + 2 * DFEAT;
    zero_kernel<<<(int)((zn + 255) / 256), 256, 0, stream>>>(wsf, zn);

    // 2. degree accumulation (edges + self loops)
    int deg_items = NEDGES + NNODES;
    deg_kernel<<<(deg_items + 255) / 256, 256, 0, stream>>>(ei, ew, deg);

    // 3. dinv = rsqrt(deg) where deg > 0
    dinv_kernel<<<(NNODES + 255) / 256, 256, 0, stream>>>(deg);

    // 4. scatter-add aggregation, one wave per edge/self-loop
    int nwaves = NEDGES + NNODES;                  // 680000 waves
    agg_kernel<<<(nwaves + 7) / 8, 256, 0, stream>>>(x, ei, ew, deg, agg);

    // 5. blended GEMM (fp32 WMMA) + ReLU + BN column stats
    gemm_bn_kernel<<<NNODES / 16, 256, 0, stream>>>(xorig, Wm, agg, out,
                                                    colsum, colsumsq);

    // 6. BN scale/shift
    bn_finalize_kernel<<<1, 128, 0, stream>>>(colsum, colsumsq, gamma, beta,
                                              scale, shift);

    // 7. normalize h in place (float4)
    long long q = (long long)NNODES * DFEAT / 4;
    bn_apply_kernel<<<(int)((q + 255) / 256), 256, 0, stream>>>(out, scale, shift);
}